// EmbeddingGNN_36197984371406
// MI455X (gfx1250) — compile-verified
//
#include <hip/hip_runtime.h>
#include <hip/hip_bf16.h>

typedef __attribute__((ext_vector_type(2))) float v2f;
typedef __attribute__((ext_vector_type(8))) float v8f;
typedef __attribute__((ext_vector_type(4))) float v4f;

// ---------------------------------------------------------------------------
// Degree / normalization kernels
// ---------------------------------------------------------------------------
__global__ void k_deg_init(float* __restrict__ deg, int n) {
  int i = blockIdx.x * blockDim.x + threadIdx.x;
  if (i < n) deg[i] = 1.0f;  // self-loop weight
}

__global__ void k_deg_accum(const long long* __restrict__ dst,
                            const float* __restrict__ ew,
                            float* __restrict__ deg, int e) {
  int i = blockIdx.x * blockDim.x + threadIdx.x;
  if (i < e) atomicAdd(&deg[(size_t)dst[i]], ew[i]);
}

__global__ void k_dinv(float* __restrict__ deg, int n) {
  int i = blockIdx.x * blockDim.x + threadIdx.x;
  if (i < n) {
    float d = deg[i];
    deg[i] = (d > 0.0f) ? rsqrtf(d) : 0.0f;  // dinv in place
  }
}

// ---------------------------------------------------------------------------
// GEMM1: h1[N,16] = x[N,256] @ W1[256,16]   (f32 WMMA 16x16x4)
// One wave per 16-row tile; W1 staged in LDS per block (4 waves).
// ---------------------------------------------------------------------------
__global__ __launch_bounds__(128) void k_gemm1(const float* __restrict__ x,
                                               const float* __restrict__ W1,
                                               float* __restrict__ h1,
                                               int ntiles) {
  __shared__ float sW[256 * 16];
  for (int i = threadIdx.x; i < 256 * 16; i += 128) sW[i] = W1[i];
  __syncthreads();

  const int wave = threadIdx.x >> 5;
  const int lane = threadIdx.x & 31;
  const int tile = blockIdx.x * 4 + wave;
  if (tile >= ntiles) return;  // wave-uniform exit: EXEC stays all-1s for WMMA

  const int m0    = tile * 16;
  const int r     = lane & 15;        // row (A) / col (B,D) within tile
  const int khalf = (lane >> 4) << 1; // 0 for lanes 0-15, 2 for lanes 16-31

  const float* __restrict__ arow = x + (size_t)(m0 + r) * 256;

  v8f acc = {};
  #pragma unroll 8
  for (int k = 0; k < 256; k += 4) {
    v2f a, b;
    // A 16x4: lane(0-15)=row M, V0/V1 = K=0,1 ; lanes 16-31 = K=2,3
    a.x = arow[k + khalf + 0];
    a.y = arow[k + khalf + 1];
    // B 4x16: V0: K=khalf row, V1: K=khalf+1 row; N = lane&15
    b.x = sW[(k + khalf + 0) * 16 + r];
    b.y = sW[(k + khalf + 1) * 16 + r];
    acc = __builtin_amdgcn_wmma_f32_16x16x4_f32(false, a, false, b,
                                                (short)0, acc, false, false);
  }

  // D 16x16: VGPR j -> row m0 + j (+8 for hi lanes), col = lane&15
  const int mrow = m0 + ((lane >> 4) << 3);
  float* __restrict__ out = h1 + (size_t)mrow * 16 + r;
  #pragma unroll
  for (int j = 0; j < 8; ++j) out[(size_t)j * 16] = acc[j];
}

// ---------------------------------------------------------------------------
// agg1[N,16] = b1 + dinv[i]^2 * h1[i]   (self-loop + bias), then edge scatter
// ---------------------------------------------------------------------------
__global__ void k_agg1_init(const float* __restrict__ h1,
                            const float* __restrict__ dinv,
                            const float* __restrict__ b1,
                            float* __restrict__ agg1, int n) {
  int t = blockIdx.x * blockDim.x + threadIdx.x;
  if (t < n * 16) {
    int i = t >> 4, c = t & 15;
    float di = dinv[i];
    agg1[t] = b1[c] + di * di * h1[t];
  }
}

__global__ void k_agg1_edges(const long long* __restrict__ src,
                             const long long* __restrict__ dst,
                             const float* __restrict__ ew,
                             const float* __restrict__ dinv,
                             const float* __restrict__ h1,
                             float* __restrict__ agg1, int e) {
  int i = blockIdx.x * blockDim.x + threadIdx.x;
  if (i >= e) return;
  size_t s = (size_t)src[i], d = (size_t)dst[i];
  float norm = dinv[s] * ew[i] * dinv[d];
  const v4f* __restrict__ hs = (const v4f*)(h1 + s * 16);
  float* __restrict__ ad = agg1 + d * 16;
  #pragma unroll
  for (int q = 0; q < 4; ++q) {
    v4f hv = hs[q];
    atomicAdd(&ad[q * 4 + 0], norm * hv.x);
    atomicAdd(&ad[q * 4 + 1], norm * hv.y);
    atomicAdd(&ad[q * 4 + 2], norm * hv.z);
    atomicAdd(&ad[q * 4 + 3], norm * hv.w);
  }
}

// ---------------------------------------------------------------------------
// GEMM2: h2[N,64] = relu(agg1[N,16]) @ W2[16,64]  (ReLU fused into A load)
// One block per 16-row tile; wave = 16-col tile (4 waves).
// ---------------------------------------------------------------------------
__global__ __launch_bounds__(128) void k_gemm2(const float* __restrict__ agg1,
                                               const float* __restrict__ W2,
                                               float* __restrict__ h2) {
  __shared__ float sW[16 * 64];
  for (int i = threadIdx.x; i < 16 * 64; i += 128) sW[i] = W2[i];
  __syncthreads();

  const int ct   = threadIdx.x >> 5;  // col tile 0..3
  const int lane = threadIdx.x & 31;
  const int m0   = blockIdx.x * 16;
  const int r     = lane & 15;
  const int khalf = (lane >> 4) << 1;

  v8f acc = {};
  #pragma unroll
  for (int k = 0; k < 16; k += 4) {
    v2f a, b;
    float a0 = agg1[(size_t)(m0 + r) * 16 + k + khalf + 0];
    float a1 = agg1[(size_t)(m0 + r) * 16 + k + khalf + 1];
    a.x = fmaxf(a0, 0.0f);  // fused ReLU
    a.y = fmaxf(a1, 0.0f);
    b.x = sW[(k + khalf + 0) * 64 + ct * 16 + r];
    b.y = sW[(k + khalf + 1) * 64 + ct * 16 + r];
    acc = __builtin_amdgcn_wmma_f32_16x16x4_f32(false, a, false, b,
                                                (short)0, acc, false, false);
  }

  const int mrow = m0 + ((lane >> 4) << 3);
  float* __restrict__ out = h2 + (size_t)mrow * 64 + ct * 16 + r;
  #pragma unroll
  for (int j = 0; j < 8; ++j) out[(size_t)j * 64] = acc[j];
}

// ---------------------------------------------------------------------------
// Output: out[N,64] = b2 + dinv^2 * h2  then edge scatter (4 threads / edge)
// ---------------------------------------------------------------------------
__global__ void k_out_init(const float* __restrict__ h2,
                           const float* __restrict__ dinv,
                           const float* __restrict__ b2,
                           float* __restrict__ outp, int n) {
  int t = blockIdx.x * blockDim.x + threadIdx.x;
  if (t < n * 64) {
    int i = t >> 6, c = t & 63;
    float di = dinv[i];
    outp[t] = b2[c] + di * di * h2[t];
  }
}

__global__ void k_out_edges(const long long* __restrict__ src,
                            const long long* __restrict__ dst,
                            const float* __restrict__ ew,
                            const float* __restrict__ dinv,
                            const float* __restrict__ h2,
                            float* __restrict__ outp, int e) {
  int t = blockIdx.x * blockDim.x + threadIdx.x;
  int i  = t >> 2;      // edge
  int cg = t & 3;       // 16-column group
  if (i >= e) return;
  size_t s = (size_t)src[i], d = (size_t)dst[i];
  float norm = dinv[s] * ew[i] * dinv[d];
  const v4f* __restrict__ hs = (const v4f*)(h2 + s * 64 + cg * 16);
  float* __restrict__ od = outp + d * 64 + cg * 16;
  #pragma unroll
  for (int q = 0; q < 4; ++q) {
    v4f hv = hs[q];
    atomicAdd(&od[q * 4 + 0], norm * hv.x);
    atomicAdd(&od[q * 4 + 1], norm * hv.y);
    atomicAdd(&od[q * 4 + 2], norm * hv.z);
    atomicAdd(&od[q * 4 + 3], norm * hv.w);
  }
}

// ---------------------------------------------------------------------------
extern "C" void kernel_launch(void* const* d_in, const int* in_sizes, int n_in,
                              void* d_out, int out_size, void* d_ws, size_t ws_size,
                              hipStream_t stream) {
  const float*     x   = (const float*)d_in[0];      // [N,256]
  const long long* ei  = (const long long*)d_in[1];  // [2,E] int64
  const float*     ew  = (const float*)d_in[2];      // [E]
  const float*     W1  = (const float*)d_in[3];      // [256,16]
  const float*     b1  = (const float*)d_in[4];      // [16]
  const float*     W2  = (const float*)d_in[5];      // [16,64]
  const float*     b2  = (const float*)d_in[6];      // [64]
  float*           out = (float*)d_out;              // [N,64]

  const int N = in_sizes[0] / 256;   // 50000
  const int E = in_sizes[2];         // 1600000
  const long long* src = ei;
  const long long* dst = ei + E;

  // Workspace layout (floats): dinv | h1[N*16] | agg1[N*16] | h2[N*64]
  float* ws   = (float*)d_ws;
  float* dinv = ws;                                  // N
  float* h1   = ws + (((size_t)N + 63) & ~(size_t)63);          // N*16
  float* agg1 = h1 + (size_t)N * 16;                 // N*16
  float* h2   = agg1 + (size_t)N * 16;               // N*64
  (void)ws_size; (void)n_in; (void)out_size;

  const int ntiles = N / 16;  // 50000 / 16 = 3125 exactly

  // 1) degree + normalization (shared by both layers)
  k_deg_init <<<(N + 255) / 256, 256, 0, stream>>>(dinv, N);
  k_deg_accum<<<(E + 255) / 256, 256, 0, stream>>>(dst, ew, dinv, E);
  k_dinv     <<<(N + 255) / 256, 256, 0, stream>>>(dinv, N);

  // 2) layer-1 dense: h1 = x @ W1   (WMMA f32 16x16x4)
  k_gemm1<<<(ntiles + 3) / 4, 128, 0, stream>>>(x, W1, h1, ntiles);

  // 3) layer-1 aggregation (bias + self-loop init, then edge atomics)
  k_agg1_init <<<((size_t)N * 16 + 255) / 256, 256, 0, stream>>>(h1, dinv, b1, agg1, N);
  k_agg1_edges<<<(E + 255) / 256, 256, 0, stream>>>(src, dst, ew, dinv, h1, agg1, E);

  // 4) layer-2 dense: h2 = relu(agg1) @ W2   (WMMA, ReLU fused)
  k_gemm2<<<ntiles, 128, 0, stream>>>(agg1, W2, h2);

  // 5) layer-2 aggregation into d_out
  k_out_init <<<((size_t)N * 64 + 255) / 256, 256, 0, stream>>>(h2, dinv, b2, out, N);
  k_out_edges<<<((size_t)E * 4 + 255) / 256, 256, 0, stream>>>(src, dst, ew, dinv, h2, out, E);
}